// Capsule_fc_80101140070966
// MI455X (gfx1250) — compile-verified
//
#include <hip/hip_runtime.h>
#include <math.h>

// Problem constants (from reference)
#define B_    256
#define I_    1152
#define J_    10
#define DIN_  8
#define DOUT_ 16
#define N_    (J_*DOUT_)     // 160 = J*DOUT; one 16-wide N tile == one output capsule j
#define CHUNK 8              // input capsules per workgroup
#define NBLK  (I_/CHUNK)     // 144 workgroups
#define WAVES 16             // one wave per 16-row batch tile (16*16 = 256 = B)
#define TPB   (WAVES*32)     // 512 threads, wave32

typedef __attribute__((ext_vector_type(2))) float v2f;
typedef __attribute__((ext_vector_type(8))) float v8f;

// One pass of fused (u-recompute via WMMA) + (b-update, softmax) + (c-weighted s accumulation).
// pass 0: c = uniform 1/J, no logit update.
// pass 1: b1 = 0 + <u, s1>;    c = softmax(b1); store b1.
// pass 2: b2 = b1 + <u, s2>;   c = softmax(b2); store b2.
__global__ __launch_bounds__(TPB)
void caps_route_pass(const float* __restrict__ x,     // [B, I, DIN]
                     const float* __restrict__ W,     // [I, J, DOUT, DIN]
                     const float* __restrict__ s_in,  // [B, J, DOUT] (squashed prev s)
                     float* __restrict__ s_raw,       // [B, J, DOUT] accumulator (pre-zeroed)
                     float* __restrict__ g_b,         // [I, J] routing logits
                     int pass)
{
    __shared__ float xs[B_ * DIN_];     // 8 KB: x[:, i, :]
    __shared__ float Ws[N_ * DIN_];     // 5 KB: W[i] flattened [j][o][d]
    __shared__ float bpart[J_];
    __shared__ float cvals[J_];

    const int tid  = threadIdx.x;
    const int wave = tid >> 5;          // batch tile: rows 16*wave .. 16*wave+15
    const int lane = tid & 31;
    const int m    = lane & 15;         // A-frag row / B-frag col / D-frag col
    const int hi   = lane >> 4;         // lane-half select
    const int i0   = blockIdx.x * CHUNK;

    float acc[J_][8];
#pragma unroll
    for (int j = 0; j < J_; ++j)
#pragma unroll
        for (int r = 0; r < 8; ++r) acc[j][r] = 0.f;

    for (int ic = 0; ic < CHUNK; ++ic) {
        const int i = i0 + ic;

        __syncthreads();   // previous iteration done with LDS
#pragma unroll 1
        for (int t = tid; t < B_*DIN_; t += TPB) {
            int b = t >> 3, k = t & 7;
            xs[t] = x[b * (I_*DIN_) + i*DIN_ + k];
        }
#pragma unroll 1
        for (int t = tid; t < N_*DIN_; t += TPB)
            Ws[t] = W[i * (N_*DIN_) + t];
        if (tid < J_) bpart[tid] = 0.f;
        __syncthreads();

        // A fragments (x tile, 16x4 per K-step): VGPR0 -> K=2*hi, VGPR1 -> K=2*hi+1
        v2f a0, a1;
        a0.x = xs[m*DIN_ + 2*hi + 0];
        a0.y = xs[m*DIN_ + 2*hi + 1];
        a1.x = xs[m*DIN_ + 4 + 2*hi + 0];
        a1.y = xs[m*DIN_ + 4 + 2*hi + 1];

        // u fragments: D[bb, j*16+o] = sum_k x[bb,k] * W[i,j,o,k]
        float ufrag[J_][8];
#pragma unroll
        for (int j = 0; j < J_; ++j) {
            // B fragments (Wt 4x16 per K-step): col n = m, row K = 2*hi (+1)
            const int wb = j * (DOUT_*DIN_) + m * DIN_;
            v2f b0, b1;
            b0.x = Ws[wb + 2*hi + 0];
            b0.y = Ws[wb + 2*hi + 1];
            b1.x = Ws[wb + 4 + 2*hi + 0];
            b1.y = Ws[wb + 4 + 2*hi + 1];
            v8f c = {0.f,0.f,0.f,0.f,0.f,0.f,0.f,0.f};
            c = __builtin_amdgcn_wmma_f32_16x16x4_f32(false, a0, false, b0,
                                                      (short)0, c, false, false);
            c = __builtin_amdgcn_wmma_f32_16x16x4_f32(false, a1, false, b1,
                                                      (short)0, c, false, false);
#pragma unroll
            for (int r = 0; r < 8; ++r) ufrag[j][r] = c[r];
        }

        // Agreement: bpart[j] = sum_b sum_d u[b,i,j,d] * s_in[b,j,d]
        if (pass > 0) {
#pragma unroll
            for (int j = 0; j < J_; ++j) {
                float dot = 0.f;
#pragma unroll
                for (int r = 0; r < 8; ++r) {
                    int bb = wave*16 + hi*8 + r;                 // D-frag row
                    dot += ufrag[j][r] * s_in[bb*N_ + j*DOUT_ + m];
                }
#pragma unroll
                for (int off = 16; off > 0; off >>= 1)
                    dot += __shfl_xor(dot, off, 32);
                if (lane == 0) atomicAdd(&bpart[j], dot);        // ds_add_f32
            }
        }
        __syncthreads();

        // Logit update + softmax over j (10 values; single thread)
        if (tid == 0) {
            float bb[J_];
            float mx = -1e30f;
#pragma unroll
            for (int j = 0; j < J_; ++j) {
                float bo = (pass == 2) ? g_b[i*J_ + j] : 0.f;    // pass1: old b == 0
                bb[j] = (pass > 0) ? (bo + bpart[j]) : 0.f;
                mx = fmaxf(mx, bb[j]);
            }
            float e[J_], sum = 0.f;
#pragma unroll
            for (int j = 0; j < J_; ++j) { e[j] = expf(bb[j] - mx); sum += e[j]; }
            float inv = 1.f / sum;
#pragma unroll
            for (int j = 0; j < J_; ++j) {
                cvals[j] = e[j] * inv;
                if (pass > 0) g_b[i*J_ + j] = bb[j];
            }
        }
        __syncthreads();

        // acc += c[j] * u (stays in VGPRs across the i-chunk)
#pragma unroll
        for (int j = 0; j < J_; ++j) {
            float cj = cvals[j];
#pragma unroll
            for (int r = 0; r < 8; ++r) acc[j][r] += cj * ufrag[j][r];
        }
    }

    // One atomic flush per chunk: s_raw[b,j,d] += sum_{i in chunk} c_i[j]*u_i
#pragma unroll
    for (int j = 0; j < J_; ++j)
#pragma unroll
        for (int r = 0; r < 8; ++r) {
            int bb = wave*16 + hi*8 + r;
            atomicAdd(&s_raw[bb*N_ + j*DOUT_ + m], acc[j][r]);   // global_atomic_add_f32
        }
}

// squash(s_raw) -> s_out, and reset s_raw to 0 for the next accumulation pass.
__global__ void caps_squash(const float* __restrict__ s_raw,
                            float* __restrict__ s_out,
                            float* __restrict__ s_reset)
{
    int row = blockIdx.x * blockDim.x + threadIdx.x;   // (b, j) pair, 2560 rows
    if (row >= B_*J_) return;
    float v[DOUT_];
    float ss = 0.f;
#pragma unroll
    for (int d = 0; d < DOUT_; ++d) {
        v[d] = s_raw[row*DOUT_ + d];
        ss += v[d]*v[d];
    }
    float l2 = sqrtf(ss);
    float sc = l2 / (1.f + ss);                        // |s| / (1 + |s|^2)
#pragma unroll
    for (int d = 0; d < DOUT_; ++d) {
        s_out[row*DOUT_ + d]   = v[d] * sc;
        s_reset[row*DOUT_ + d] = 0.f;
    }
}

__global__ void caps_zero(float* __restrict__ p, int n)
{
    int t = blockIdx.x * blockDim.x + threadIdx.x;
    if (t < n) p[t] = 0.f;
}

extern "C" void kernel_launch(void* const* d_in, const int* in_sizes, int n_in,
                              void* d_out, int out_size, void* d_ws, size_t ws_size,
                              hipStream_t stream)
{
    (void)in_sizes; (void)n_in; (void)out_size; (void)ws_size;
    const float* x = (const float*)d_in[0];   // [256, 1152, 8]
    const float* W = (const float*)d_in[1];   // [1152, 10, 16, 8]
    float* out   = (float*)d_out;             // [256, 10, 16]

    float* s_raw = (float*)d_ws;              // 40960 f32
    float* s_buf = s_raw + B_*N_;             // 40960 f32
    float* g_b   = s_buf + B_*N_;             // 11520 f32
    // total workspace: ~365 KB

    caps_zero<<<(B_*N_ + 255)/256, 256, 0, stream>>>(s_raw, B_*N_);

    // Iteration 1: c uniform
    caps_route_pass<<<NBLK, TPB, 0, stream>>>(x, W, s_buf, s_raw, g_b, 0);
    caps_squash<<<(B_*J_ + 127)/128, 128, 0, stream>>>(s_raw, s_buf, s_raw);
    // Iteration 2: b1 = <u, s1>
    caps_route_pass<<<NBLK, TPB, 0, stream>>>(x, W, s_buf, s_raw, g_b, 1);
    caps_squash<<<(B_*J_ + 127)/128, 128, 0, stream>>>(s_raw, s_buf, s_raw);
    // Iteration 3: b2 = b1 + <u, s2>; final s -> d_out
    caps_route_pass<<<NBLK, TPB, 0, stream>>>(x, W, s_buf, s_raw, g_b, 2);
    caps_squash<<<(B_*J_ + 127)/128, 128, 0, stream>>>(s_raw, out, s_raw);
}